// AttentiveConvMoleculeModel_57836029608282
// MI455X (gfx1250) — compile-verified
//
#include <hip/hip_runtime.h>
#include <cmath>

#define NN   50000
#define EE   800000
#define FIN  32
#define EDIM 10
#define GG   512
#define CC   64

typedef __attribute__((ext_vector_type(2))) float v2f;
typedef __attribute__((ext_vector_type(8))) float v8f;

static inline int cdiv(long a, long b) { return (int)((a + b - 1) / b); }

// ---------------------------------------------------------------- utilities
__global__ void __launch_bounds__(256) fill_kernel(float* p, float v, size_t n) {
  size_t i = (size_t)blockIdx.x * blockDim.x + threadIdx.x;
  if (i < n) p[i] = v;
}

__device__ __forceinline__ void atomicMaxF(float* a, float v) {
  // monotonic mapping trick; buffer initialized to -inf (0xFF800000)
  if (v >= 0.f) atomicMax((int*)a, __float_as_int(v));
  else          atomicMin((unsigned int*)a, (unsigned int)__float_as_int(v));
}

// ---------------------------------------------------------------- WMMA GEMM
// C[M x NC] = A[M x K] @ B[K x NC], f32, one wave per 16x16 tile.
// A frag (16x4): lane m=lane&15, VGPR0 = K=k0+2*half, VGPR1 = K=k0+2*half+1
// B frag (4x16): symmetric striping; C/D: VGPR r -> M = r + 8*half, N = lane&15
__global__ void __launch_bounds__(32) gemm16_wmma(
    const float* __restrict__ A, const float* __restrict__ B,
    float* __restrict__ C, int K, int NC) {
  const int mBase = blockIdx.x * 16;
  const int nBase = blockIdx.y * 16;
  const int lane  = threadIdx.x;
  const int half  = lane >> 4;
  const int l16   = lane & 15;
  const float* Arow = A + (size_t)(mBase + l16) * K;
  const float* Bcol = B + nBase + l16;
  v8f acc = {};
#pragma unroll 4
  for (int k0 = 0; k0 < K; k0 += 4) {
    const int ka = k0 + 2 * half;
    v2f a = { Arow[ka], Arow[ka + 1] };
    v2f b = { Bcol[(size_t)ka * NC], Bcol[(size_t)(ka + 1) * NC] };
    acc = __builtin_amdgcn_wmma_f32_16x16x4_f32(false, a, false, b,
                                                (short)0, acc, false, false);
  }
  float* Crow = C + nBase + l16;
#pragma unroll
  for (int r = 0; r < 8; ++r)
    Crow[(size_t)(mBase + r + 8 * half) * NC] = acc[r];
}

// ---------------------------------------------------------------- graph prep
__global__ void __launch_bounds__(256) deg_kernel(const int* __restrict__ dst,
                                                  float* deg, int E) {
  int e = blockIdx.x * blockDim.x + threadIdx.x;
  if (e < E) atomicAdd(&deg[dst[e]], 1.0f);
}

__global__ void __launch_bounds__(256) loopea_acc(const int* __restrict__ dst,
                                                  const float* __restrict__ ea,
                                                  float* loopea, int E) {
  int e = blockIdx.x * blockDim.x + threadIdx.x;
  if (e >= E) return;
  int d = dst[e];
#pragma unroll
  for (int i = 0; i < EDIM; ++i)
    atomicAdd(&loopea[(size_t)d * EDIM + i], ea[(size_t)e * EDIM + i]);
}

__global__ void __launch_bounds__(256) loopea_div(float* loopea,
                                                  const float* __restrict__ deg) {
  int i = blockIdx.x * blockDim.x + threadIdx.x;
  if (i < NN * EDIM) loopea[i] /= fmaxf(deg[i / EDIM], 1.0f);
}

__global__ void __launch_bounds__(256) cnt_kernel(const int* __restrict__ batch,
                                                  float* cnt) {
  int n = blockIdx.x * blockDim.x + threadIdx.x;
  if (n < NN) atomicAdd(&cnt[batch[n]], 1.0f);
}

// ---------------------------------------------------------------- attention
// Me[i][h] = sum_c We[i, h*C + c] * ae[h, c]   (collapses the edge GEMM)
__global__ void __launch_bounds__(64) me_kernel(const float* __restrict__ We,
                                                const float* __restrict__ ae,
                                                float* Me, int H) {
  int t = threadIdx.x;
  if (t >= EDIM * H) return;
  int i = t / H, h = t % H;
  float s = 0.f;
  const float* w = We + (size_t)i * H * CC + (size_t)h * CC;
  const float* a = ae + (size_t)h * CC;
#pragma unroll 8
  for (int c = 0; c < CC; ++c) s += w[c] * a[c];
  Me[i * H + h] = s;
}

__global__ void __launch_bounds__(256) attn_node(const float* __restrict__ xw,
                                                 const float* __restrict__ a_s,
                                                 const float* __restrict__ a_d,
                                                 float* asrc, float* adst, int H) {
  int idx = blockIdx.x * blockDim.x + threadIdx.x;
  if (idx >= NN * H) return;
  int n = idx / H, h = idx % H;
  const float* row = xw + (size_t)n * H * CC + (size_t)h * CC;
  const float* vs = a_s + (size_t)h * CC;
  const float* vd = a_d + (size_t)h * CC;
  float s = 0.f, d = 0.f;
#pragma unroll 8
  for (int c = 0; c < CC; ++c) { float v = row[c]; s += v * vs[c]; d += v * vd[c]; }
  asrc[idx] = s; adst[idx] = d;
}

__global__ void __launch_bounds__(256) edge_logits(
    const int* __restrict__ src, const int* __restrict__ dst,
    const float* __restrict__ ea, const float* __restrict__ asrc,
    const float* __restrict__ adst, const float* __restrict__ Me,
    float* elog, float* m, int H) {
  int e = blockIdx.x * blockDim.x + threadIdx.x;
  if (e >= EE) return;
  int s = src[e], d = dst[e];
  float eav[EDIM];
#pragma unroll
  for (int i = 0; i < EDIM; ++i) eav[i] = ea[(size_t)e * EDIM + i];
  for (int h = 0; h < H; ++h) {
    float aed = 0.f;
#pragma unroll
    for (int i = 0; i < EDIM; ++i) aed += eav[i] * Me[i * H + h];
    float z = asrc[s * H + h] + adst[d * H + h] + aed;
    z = z > 0.f ? z : 0.2f * z;
    elog[(size_t)e * H + h] = z;
    atomicMaxF(&m[d * H + h], z);
  }
}

__global__ void __launch_bounds__(256) loop_logits(
    const float* __restrict__ loopea, const float* __restrict__ asrc,
    const float* __restrict__ adst, const float* __restrict__ Me,
    float* llog, float* m, int H) {
  int n = blockIdx.x * blockDim.x + threadIdx.x;
  if (n >= NN) return;
  for (int h = 0; h < H; ++h) {
    float aed = 0.f;
#pragma unroll
    for (int i = 0; i < EDIM; ++i) aed += loopea[(size_t)n * EDIM + i] * Me[i * H + h];
    float z = asrc[n * H + h] + adst[n * H + h] + aed;
    z = z > 0.f ? z : 0.2f * z;
    llog[n * H + h] = z;
    atomicMaxF(&m[n * H + h], z);
  }
}

__global__ void __launch_bounds__(256) edge_exp(const int* __restrict__ dst,
                                                float* elog, const float* __restrict__ m,
                                                float* dnm, int H) {
  int e = blockIdx.x * blockDim.x + threadIdx.x;
  if (e >= EE) return;
  int d = dst[e];
  for (int h = 0; h < H; ++h) {
    float p = expf(elog[(size_t)e * H + h] - m[d * H + h]);
    elog[(size_t)e * H + h] = p;
    atomicAdd(&dnm[d * H + h], p);
  }
}

__global__ void __launch_bounds__(256) loop_exp(float* llog, const float* __restrict__ m,
                                                float* dnm, int H) {
  int n = blockIdx.x * blockDim.x + threadIdx.x;
  if (n >= NN) return;
  for (int h = 0; h < H; ++h) {
    float p = expf(llog[n * H + h] - m[n * H + h]);
    llog[n * H + h] = p;
    atomicAdd(&dnm[n * H + h], p);
  }
}

// one wave32 per edge; gathers hit L2 (xw is 51MB << 192MB L2)
__global__ void __launch_bounds__(256) scatter_edges(
    const int* __restrict__ src, const int* __restrict__ dst,
    const float* __restrict__ elog, const float* __restrict__ dnm,
    const float* __restrict__ xw, float* outacc, int H) {
  int wave = blockIdx.x * (blockDim.x >> 5) + (threadIdx.x >> 5);
  int lane = threadIdx.x & 31;
  if (wave >= EE) return;
  int s = src[wave], d = dst[wave];
  const int HC = H * CC;
  float alpha[4];
  for (int h = 0; h < H; ++h)
    alpha[h] = elog[(size_t)wave * H + h] / dnm[d * H + h];
  const float* xs = xw + (size_t)s * HC;
  float* od = outacc + (size_t)d * HC;
  for (int i = lane; i < HC; i += 32)
    atomicAdd(&od[i], alpha[i >> 6] * xs[i]);
}

__global__ void __launch_bounds__(256) scatter_loops(
    const float* __restrict__ llog, const float* __restrict__ dnm,
    const float* __restrict__ xw, float* outacc, int H) {
  int wave = blockIdx.x * (blockDim.x >> 5) + (threadIdx.x >> 5);
  int lane = threadIdx.x & 31;
  if (wave >= NN) return;
  const int HC = H * CC;
  float alpha[4];
  for (int h = 0; h < H; ++h)
    alpha[h] = llog[wave * H + h] / dnm[wave * H + h];
  const float* xs = xw + (size_t)wave * HC;
  float* od = outacc + (size_t)wave * HC;
  for (int i = lane; i < HC; i += 32)   // unique node per wave -> plain add
    od[i] += alpha[i >> 6] * xs[i];
}

// ---------------------------------------------------------------- BN + act
__global__ void __launch_bounds__(256) head_mean_stats(
    const float* __restrict__ outacc, const float* __restrict__ bias,
    float* y, float* bnsum, float* bnsq, int H) {
  int t = blockIdx.x * blockDim.x + threadIdx.x;
  int c = t & 63;
  int r0 = t >> 6;
  int rstride = (gridDim.x * blockDim.x) >> 6;
  const float invH = 1.0f / (float)H;
  float s = 0.f, sq = 0.f;
  for (int n = r0; n < NN; n += rstride) {
    float v = 0.f;
    for (int h = 0; h < H; ++h) v += outacc[(size_t)n * H * CC + (size_t)h * CC + c];
    v = v * invH + bias[c];
    y[(size_t)n * CC + c] = v;
    s += v; sq += v * v;
  }
  atomicAdd(&bnsum[c], s);
  atomicAdd(&bnsq[c], sq);
}

__global__ void __launch_bounds__(64) bn_finalize(
    const float* __restrict__ bnsum, const float* __restrict__ bnsq,
    const float* __restrict__ g, const float* __restrict__ be,
    float* scale, float* shift) {
  int c = threadIdx.x;
  float mu  = bnsum[c] / (float)NN;
  float var = bnsq[c] / (float)NN - mu * mu;
  float rs  = rsqrtf(var + 1e-5f) * g[c];
  scale[c] = rs;
  shift[c] = be[c] - mu * rs;
}

__global__ void __launch_bounds__(256) apply_act(
    const float* __restrict__ y, const float* __restrict__ scale,
    const float* __restrict__ shift, float* hout, int act) {
  int i = blockIdx.x * blockDim.x + threadIdx.x;
  if (i >= NN * CC) return;
  int c = i & 63;
  float v = y[i] * scale[c] + shift[c];
  hout[i] = (act == 0) ? fmaxf(v, 0.f) : (v > 0.f ? v : 0.01f * v);
}

// ---------------------------------------------------------------- pool + MLP
__global__ void __launch_bounds__(256) pool_acc(const float* __restrict__ h,
                                                const int* __restrict__ batch,
                                                float* pooled) {
  int i = blockIdx.x * blockDim.x + threadIdx.x;
  if (i >= NN * CC) return;
  int n = i >> 6, c = i & 63;
  atomicAdd(&pooled[(size_t)batch[n] * CC + c], h[i]);
}

__global__ void __launch_bounds__(64) mlp_kernel(
    const float* __restrict__ pooled, const float* __restrict__ cnt,
    const float* __restrict__ fw1, const float* __restrict__ fb1,
    const float* __restrict__ fw2, const float* __restrict__ fb2,
    float* out) {
  int g = blockIdx.x, t = threadIdx.x;
  __shared__ float row[CC];
  __shared__ float red[CC];
  row[t] = pooled[(size_t)g * CC + t] / fmaxf(cnt[g], 1.0f);
  __syncthreads();
  float hid = fb1[t];
  for (int c = 0; c < CC; ++c) hid += row[c] * fw1[c * CC + t];
  hid = fmaxf(hid, 0.f);
  red[t] = hid * fw2[t];
  __syncthreads();
  for (int s = 32; s > 0; s >>= 1) {
    if (t < s) red[t] += red[t + s];
    __syncthreads();
  }
  if (t == 0) out[g] = red[0] + fb2[0];
}

// ---------------------------------------------------------------- driver
static void launch_layer(const float* hin, int F, int H,
    const float* Wl, const float* asl, const float* adl,
    const float* Wel, const float* ael, const float* bl,
    const float* gl, const float* bel,
    const int* src, const int* dst, const float* ea, const float* loopea,
    float* hout, int act,
    float* xw, float* outacc, float* y, float* asrc, float* adst,
    float* elog, float* llog, float* mbuf, float* dnm,
    float* Me, float* bnsum, float* bnsq, float* scale, float* shift,
    hipStream_t stream) {
  const int HC = H * CC;
  me_kernel<<<1, 64, 0, stream>>>(Wel, ael, Me, H);
  dim3 gg(NN / 16, HC / 16);
  gemm16_wmma<<<gg, 32, 0, stream>>>(hin, Wl, xw, F, HC);
  attn_node<<<cdiv((long)NN * H, 256), 256, 0, stream>>>(xw, asl, adl, asrc, adst, H);
  fill_kernel<<<cdiv((long)NN * H, 256), 256, 0, stream>>>(mbuf, -INFINITY, (size_t)NN * H);
  fill_kernel<<<cdiv((long)NN * H, 256), 256, 0, stream>>>(dnm, 0.f, (size_t)NN * H);
  fill_kernel<<<cdiv((long)NN * HC, 256), 256, 0, stream>>>(outacc, 0.f, (size_t)NN * HC);
  edge_logits<<<cdiv(EE, 256), 256, 0, stream>>>(src, dst, ea, asrc, adst, Me, elog, mbuf, H);
  loop_logits<<<cdiv(NN, 256), 256, 0, stream>>>(loopea, asrc, adst, Me, llog, mbuf, H);
  edge_exp<<<cdiv(EE, 256), 256, 0, stream>>>(dst, elog, mbuf, dnm, H);
  loop_exp<<<cdiv(NN, 256), 256, 0, stream>>>(llog, mbuf, dnm, H);
  scatter_edges<<<cdiv(EE, 8), 256, 0, stream>>>(src, dst, elog, dnm, xw, outacc, H);
  scatter_loops<<<cdiv(NN, 8), 256, 0, stream>>>(llog, dnm, xw, outacc, H);
  fill_kernel<<<1, 256, 0, stream>>>(bnsum, 0.f, 128);   // bnsum+bnsq contiguous
  head_mean_stats<<<128, 256, 0, stream>>>(outacc, bl, y, bnsum, bnsq, H);
  bn_finalize<<<1, 64, 0, stream>>>(bnsum, bnsq, gl, bel, scale, shift);
  apply_act<<<cdiv((long)NN * CC, 256), 256, 0, stream>>>(y, scale, shift, hout, act);
}

extern "C" void kernel_launch(void* const* d_in, const int* in_sizes, int n_in,
                              void* d_out, int out_size, void* d_ws, size_t ws_size,
                              hipStream_t stream) {
  (void)in_sizes; (void)n_in; (void)out_size; (void)ws_size;
  const float* x     = (const float*)d_in[0];
  const int*   ei    = (const int*)d_in[1];
  const float* ea    = (const float*)d_in[2];
  const int*   batch = (const int*)d_in[3];
  const float* w[3]  = {(const float*)d_in[4],  (const float*)d_in[12], (const float*)d_in[20]};
  const float* as_[3]= {(const float*)d_in[5],  (const float*)d_in[13], (const float*)d_in[21]};
  const float* ad_[3]= {(const float*)d_in[6],  (const float*)d_in[14], (const float*)d_in[22]};
  const float* we[3] = {(const float*)d_in[7],  (const float*)d_in[15], (const float*)d_in[23]};
  const float* aeP[3]= {(const float*)d_in[8],  (const float*)d_in[16], (const float*)d_in[24]};
  const float* b_[3] = {(const float*)d_in[9],  (const float*)d_in[17], (const float*)d_in[25]};
  const float* g_[3] = {(const float*)d_in[10], (const float*)d_in[18], (const float*)d_in[26]};
  const float* be_[3]= {(const float*)d_in[11], (const float*)d_in[19], (const float*)d_in[27]};
  const float* fw1 = (const float*)d_in[28];
  const float* fb1 = (const float*)d_in[29];
  const float* fw2 = (const float*)d_in[30];
  const float* fb2 = (const float*)d_in[31];
  const int* src = ei;
  const int* dst = ei + EE;
  float* out = (float*)d_out;

  // workspace layout (floats)
  float* W0 = (float*)d_ws;
  size_t o = 0;
  auto alloc = [&](size_t n) { float* p = W0 + o; o += n; return p; };
  float* xw     = alloc((size_t)NN * 256);
  float* outacc = alloc((size_t)NN * 256);
  float* hA     = alloc((size_t)NN * CC);
  float* hB     = alloc((size_t)NN * CC);
  float* y      = alloc((size_t)NN * CC);
  float* asrc   = alloc((size_t)NN * 4);
  float* adst   = alloc((size_t)NN * 4);
  float* elog   = alloc((size_t)EE * 4);
  float* llog   = alloc((size_t)NN * 4);
  float* mbuf   = alloc((size_t)NN * 4);
  float* dnm    = alloc((size_t)NN * 4);
  float* loopea = alloc((size_t)NN * EDIM);
  float* deg    = alloc((size_t)NN);
  float* Me     = alloc(64);
  float* bnsum  = alloc(64);
  float* bnsq   = alloc(64);   // contiguous with bnsum (filled together)
  float* scale  = alloc(64);
  float* shift  = alloc(64);
  float* pooled = alloc((size_t)GG * CC);
  float* cnt    = alloc((size_t)GG);

  // graph-invariant precompute (re-done every call; deterministic)
  fill_kernel<<<cdiv(NN, 256), 256, 0, stream>>>(deg, 0.f, (size_t)NN);
  fill_kernel<<<cdiv((long)NN * EDIM, 256), 256, 0, stream>>>(loopea, 0.f, (size_t)NN * EDIM);
  fill_kernel<<<cdiv((long)GG * CC, 256), 256, 0, stream>>>(pooled, 0.f, (size_t)GG * CC);
  fill_kernel<<<cdiv(GG, 256), 256, 0, stream>>>(cnt, 0.f, (size_t)GG);
  deg_kernel<<<cdiv(EE, 256), 256, 0, stream>>>(dst, deg, EE);
  loopea_acc<<<cdiv(EE, 256), 256, 0, stream>>>(dst, ea, loopea, EE);
  loopea_div<<<cdiv((long)NN * EDIM, 256), 256, 0, stream>>>(loopea, deg);
  cnt_kernel<<<cdiv(NN, 256), 256, 0, stream>>>(batch, cnt);

  // layer 1: F=32, H=4, relu
  launch_layer(x, FIN, 4, w[0], as_[0], ad_[0], we[0], aeP[0], b_[0], g_[0], be_[0],
               src, dst, ea, loopea, hA, 0,
               xw, outacc, y, asrc, adst, elog, llog, mbuf, dnm,
               Me, bnsum, bnsq, scale, shift, stream);
  // layer 2: F=64, H=2, relu
  launch_layer(hA, CC, 2, w[1], as_[1], ad_[1], we[1], aeP[1], b_[1], g_[1], be_[1],
               src, dst, ea, loopea, hB, 0,
               xw, outacc, y, asrc, adst, elog, llog, mbuf, dnm,
               Me, bnsum, bnsq, scale, shift, stream);
  // layer 3: F=64, H=4, leaky_relu(0.01)
  launch_layer(hB, CC, 4, w[2], as_[2], ad_[2], we[2], aeP[2], b_[2], g_[2], be_[2],
               src, dst, ea, loopea, hA, 1,
               xw, outacc, y, asrc, adst, elog, llog, mbuf, dnm,
               Me, bnsum, bnsq, scale, shift, stream);

  // global mean pool + MLP head
  pool_acc<<<cdiv((long)NN * CC, 256), 256, 0, stream>>>(hA, batch, pooled);
  mlp_kernel<<<GG, 64, 0, stream>>>(pooled, cnt, fw1, fb1, fw2, fb2, out);
}